// SplitterBasisComination_42545946034647
// MI455X (gfx1250) — compile-verified
//
#include <hip/hip_runtime.h>
#include <cstdint>

// Problem constants (fixed by the reference):
// x: [B=16, C=8, T=4096, L=512] fp32 ; w: [SEG=128] fp32
// out: [B, C, K=32, L=512] fp32 ; out[slice, l] = sum_s w[s]*x[slice, s, l]
#define SEG_   128
#define LDIM_  512
#define NSLICE 4096          // B*C*K
#define KCHUNKS (SEG_ / 4)   // 32 WMMA K=4 chunks

typedef float v2f __attribute__((ext_vector_type(2)));
typedef float v8f __attribute__((ext_vector_type(8)));

// One wave computes out[slice, l0 .. l0+15] via 32 chained
// V_WMMA_F32_16X16X4_F32 ops:
//   A (16x4)  = w[4c .. 4c+3] replicated down all 16 rows
//   B (4x16)  = x[slice, 4c .. 4c+3, l0 .. l0+15]
//   D (16x16) = every row m equals out[l0 .. l0+15]; keep row M=0.
__global__ __launch_bounds__(256)
void splitter_wmma_kernel(const float* __restrict__ x,
                          const float* __restrict__ w,
                          float* __restrict__ out)
{
    __shared__ __align__(16) float wlds[SEG_];

    // ---- Stage w (512 B) into LDS with the async-to-LDS path (ASYNCcnt). ----
    // Wave 0: 32 lanes x 16 B = 512 B in one async b128 per lane.
    if (threadIdx.x < 32) {
        uint32_t lds_off = (uint32_t)(uintptr_t)(&wlds[threadIdx.x * 4]);
        const float* gsrc = w + threadIdx.x * 4;
        asm volatile("global_load_async_to_lds_b128 %0, %1, off"
                     :: "v"(lds_off), "v"(gsrc) : "memory");
    }
    asm volatile("s_wait_asynccnt 0" ::: "memory");
    __syncthreads();

    const int lane = threadIdx.x & 31;
    const int wave = threadIdx.x >> 5;
    const int col  = lane & 15;     // N index within the 16-wide l tile
    const int half = lane >> 4;     // 0: rows {0,1} of pair, 1: rows {2,3}

    const int slice = blockIdx.x >> 2;                          // 0..4095
    const int l0    = ((blockIdx.x & 3) << 7) + (wave << 4);    // 0..496

    const float* xb = x + (size_t)slice * (SEG_ * LDIM_) + l0 + col;

    v8f acc = {0.f, 0.f, 0.f, 0.f, 0.f, 0.f, 0.f, 0.f};

#pragma unroll
    for (int c = 0; c < KCHUNKS; ++c) {
        const int r0 = 4 * c + 2 * half;    // this lane supplies rows r0, r0+1

        // A-matrix lane data: VGPR0 = w[K=0]/w[K=2], VGPR1 = w[K=1]/w[K=3]
        v2f a;
        a.x = wlds[r0];
        a.y = wlds[r0 + 1];

        // B-matrix lane data: VGPR0 = X[K=0 or 2, col], VGPR1 = X[K=1 or 3, col]
        v2f b;
        b.x = xb[(size_t)r0 * LDIM_];
        b.y = xb[(size_t)(r0 + 1) * LDIM_];

        // 8 args: (neg_a, A, neg_b, B, c_mod, C, reuse_a, reuse_b)
        acc = __builtin_amdgcn_wmma_f32_16x16x4_f32(
            false, a, false, b, (short)0, acc, false, false);
    }

    // D row M=0 lives in acc[0] of lanes 0..15 (N = lane).
    if (lane < 16) {
        out[(size_t)slice * LDIM_ + l0 + col] = acc[0];
    }
}

extern "C" void kernel_launch(void* const* d_in, const int* in_sizes, int n_in,
                              void* d_out, int out_size, void* d_ws, size_t ws_size,
                              hipStream_t stream) {
    (void)in_sizes; (void)n_in; (void)out_size; (void)d_ws; (void)ws_size;
    const float* x = (const float*)d_in[0];
    const float* w = (const float*)d_in[1];
    float* out = (float*)d_out;

    dim3 grid(NSLICE * 4);   // 4 blocks of 128 l-columns per slice
    dim3 block(256);         // 8 waves (wave32)
    splitter_wmma_kernel<<<grid, block, 0, stream>>>(x, w, out);
}